// MAGICModel_12421045420438
// MI455X (gfx1250) — compile-verified
//
#include <hip/hip_runtime.h>
#include <hip/hip_bf16.h>

typedef __attribute__((ext_vector_type(2))) float v2f;
typedef __attribute__((ext_vector_type(8))) float v8f;

#define D_NODE 64
#define D_EDGE 32
#define NEG_SLOPE 0.2f
#define EDGES_PER_BLOCK 128   // 8 waves * 16 edges
#define NODES_PER_BLOCK 128   // 8 waves * 16 nodes

// ---------------------------------------------------------------------------
// Zero the aggregation buffer (graph-capture-safe, no memset API needed)
// ---------------------------------------------------------------------------
__global__ __launch_bounds__(256) void magic_zero_kernel(float* __restrict__ p,
                                                         unsigned long long n) {
    unsigned long long i = (unsigned long long)blockIdx.x * blockDim.x + threadIdx.x;
    if (i < n) p[i] = 0.0f;
}

// ---------------------------------------------------------------------------
// Edge kernel: ea = edge_attr @ Wa^T + ba  (WMMA f32 16x16x4)
//              m  = relu(h[src] + ea)
//              agg[dst] += m               (global f32 atomics, L2-resident)
// One wave32 handles 16 edges x 64 output dims (4 N-tiles, K=32 in 8 steps).
// ---------------------------------------------------------------------------
__global__ __launch_bounds__(256) void magic_edge_kernel(
    const float* __restrict__ h,          // [N,64] node features
    const int*   __restrict__ src,        // [E]
    const int*   __restrict__ dst,        // [E]
    const float* __restrict__ edge_attr,  // [E,32]
    const float* __restrict__ Wa,         // [64,32]
    const float* __restrict__ ba,         // [64]
    float*       __restrict__ agg,        // [N,64] (pre-zeroed)
    int n_edges)
{
    __shared__ int s_src[EDGES_PER_BLOCK];
    __shared__ int s_dst[EDGES_PER_BLOCK];

    const int tid  = threadIdx.x;
    const int wave = tid >> 5;
    const int lane = tid & 31;
    const int half = lane >> 4;    // 0: lanes 0-15, 1: lanes 16-31
    const int l16  = lane & 15;

    const int ebase_blk = blockIdx.x * EDGES_PER_BLOCK;

    // stage this block's edge indices into LDS
    for (int i = tid; i < EDGES_PER_BLOCK; i += 256) {
        int e = ebase_blk + i;
        s_src[i] = (e < n_edges) ? src[e] : 0;
        s_dst[i] = (e < n_edges) ? dst[e] : -1;
    }
    __syncthreads();

    const int ebase = ebase_blk + wave * 16;
    if (ebase >= n_edges) return;   // wave-uniform exit, EXEC stays all-1 for WMMA

    // A-matrix lane addressing (16x4 f32 frag): row M = l16, K-pair = 2*half
    int e_a = ebase + l16;
    if (e_a >= n_edges) e_a = n_edges - 1;   // clamp: compute garbage, store guarded
    const float* arow = edge_attr + (size_t)e_a * D_EDGE;

    v8f c[4] = {v8f{}, v8f{}, v8f{}, v8f{}};

    #pragma unroll
    for (int kt = 0; kt < D_EDGE / 4; ++kt) {
        const int k0 = kt * 4 + 2 * half;
        v2f a;
        a.x = arow[k0];
        a.y = arow[k0 + 1];
        #pragma unroll
        for (int nt = 0; nt < 4; ++nt) {
            const int n = nt * 16 + l16;
            v2f bf;                              // B[k][n] = Wa[n][k]
            bf.x = Wa[n * D_EDGE + k0];
            bf.y = Wa[n * D_EDGE + k0 + 1];
            c[nt] = __builtin_amdgcn_wmma_f32_16x16x4_f32(
                false, a, false, bf, (short)0, c[nt], false, false);
        }
    }

    // epilogue: bias + gather + relu + scatter-add
    #pragma unroll
    for (int nt = 0; nt < 4; ++nt) {
        const int d = nt * 16 + l16;
        const float bias = ba[d];
        #pragma unroll
        for (int r = 0; r < 8; ++r) {
            const int er = r + 8 * half;         // C/D layout: M = r + 8*(lane/16)
            const int e  = ebase + er;
            if (e < n_edges) {
                const int sidx = s_src[wave * 16 + er];
                const int didx = s_dst[wave * 16 + er];
                float v = c[nt][r] + bias + h[(size_t)sidx * D_NODE + d];
                v = v > 0.0f ? v : 0.0f;
                atomicAdd(&agg[(size_t)didx * D_NODE + d], v);
            }
        }
    }
}

// ---------------------------------------------------------------------------
// Node kernel: z = concat(agg, (1+eps)*h) @ W^T + b  (K=128 via WMMA f32)
//              out = final_relu ? relu(z) : leaky_relu(z, 0.2)
// One wave32 handles 16 nodes x 64 output dims.
// ---------------------------------------------------------------------------
__global__ __launch_bounds__(256) void magic_node_kernel(
    const float* __restrict__ agg,   // [N,64]
    const float* __restrict__ h,     // [N,64]
    const float* __restrict__ W,     // [64,128]
    const float* __restrict__ b,     // [64]
    const float* __restrict__ eps,   // [1]
    float*       __restrict__ out,   // [N,64]
    int n_nodes, int final_relu)
{
    const int tid  = threadIdx.x;
    const int wave = tid >> 5;
    const int lane = tid & 31;
    const int half = lane >> 4;
    const int l16  = lane & 15;

    const int nbase = (blockIdx.x * 8 + wave) * 16;
    if (nbase >= n_nodes) return;    // wave-uniform

    int row = nbase + l16;
    if (row >= n_nodes) row = n_nodes - 1;   // clamped A loads; stores guarded

    const float scale = 1.0f + eps[0];
    const float* aggrow = agg + (size_t)row * D_NODE;
    const float* hrow   = h   + (size_t)row * D_NODE;

    v8f c[4] = {v8f{}, v8f{}, v8f{}, v8f{}};

    // K = 0..63 : aggregated messages
    #pragma unroll
    for (int kt = 0; kt < 16; ++kt) {
        const int k0 = kt * 4 + 2 * half;
        v2f a;
        a.x = aggrow[k0];
        a.y = aggrow[k0 + 1];
        #pragma unroll
        for (int nt = 0; nt < 4; ++nt) {
            const int n = nt * 16 + l16;
            v2f bf;                           // B[k][n] = W[n][k]
            bf.x = W[n * 128 + k0];
            bf.y = W[n * 128 + k0 + 1];
            c[nt] = __builtin_amdgcn_wmma_f32_16x16x4_f32(
                false, a, false, bf, (short)0, c[nt], false, false);
        }
    }
    // K = 64..127 : (1+eps) * h residual
    #pragma unroll
    for (int kt = 0; kt < 16; ++kt) {
        const int k0 = kt * 4 + 2 * half;
        v2f a;
        a.x = scale * hrow[k0];
        a.y = scale * hrow[k0 + 1];
        #pragma unroll
        for (int nt = 0; nt < 4; ++nt) {
            const int n = nt * 16 + l16;
            v2f bf;
            bf.x = W[n * 128 + 64 + k0];
            bf.y = W[n * 128 + 64 + k0 + 1];
            c[nt] = __builtin_amdgcn_wmma_f32_16x16x4_f32(
                false, a, false, bf, (short)0, c[nt], false, false);
        }
    }

    // epilogue: bias + activation + store
    #pragma unroll
    for (int nt = 0; nt < 4; ++nt) {
        const int d = nt * 16 + l16;
        const float bias = b[d];
        #pragma unroll
        for (int r = 0; r < 8; ++r) {
            const int node = nbase + r + 8 * half;
            if (node < n_nodes) {
                float z = c[nt][r] + bias;
                float v;
                if (final_relu) v = z > 0.0f ? z : 0.0f;             // relu(leaky(z)) == relu(z)
                else            v = z > 0.0f ? z : NEG_SLOPE * z;
                out[(size_t)node * D_NODE + d] = v;
            }
        }
    }
}

// ---------------------------------------------------------------------------
// Host launcher
// ---------------------------------------------------------------------------
extern "C" void kernel_launch(void* const* d_in, const int* in_sizes, int n_in,
                              void* d_out, int out_size, void* d_ws, size_t ws_size,
                              hipStream_t stream) {
    (void)n_in; (void)out_size; (void)ws_size;

    const float* x         = (const float*)d_in[0];
    const int*   edge_idx  = (const int*)d_in[1];     // [2, E] int32
    const float* edge_attr = (const float*)d_in[2];
    const float* Wa1 = (const float*)d_in[3];
    const float* ba1 = (const float*)d_in[4];
    const float* W1  = (const float*)d_in[5];
    const float* b1  = (const float*)d_in[6];
    const float* e1  = (const float*)d_in[7];
    const float* Wa2 = (const float*)d_in[8];
    const float* ba2 = (const float*)d_in[9];
    const float* W2  = (const float*)d_in[10];
    const float* b2  = (const float*)d_in[11];
    const float* e2  = (const float*)d_in[12];

    const int n_nodes = in_sizes[0] / D_NODE;
    const int n_edges = in_sizes[1] / 2;
    const int* src = edge_idx;
    const int* dst = edge_idx + n_edges;

    const unsigned long long feat_elems = (unsigned long long)n_nodes * D_NODE;
    float* agg = (float*)d_ws;                 // [N,64], reused by both layers
    float* h1  = agg + feat_elems;             // [N,64] layer-1 output
    float* out = (float*)d_out;                // [N,64] final output

    const int zero_blocks = (int)((feat_elems + 255) / 256);
    const int edge_blocks = (n_edges + EDGES_PER_BLOCK - 1) / EDGES_PER_BLOCK;
    const int node_blocks = (n_nodes + NODES_PER_BLOCK - 1) / NODES_PER_BLOCK;

    // ---- layer 1 ----
    magic_zero_kernel<<<zero_blocks, 256, 0, stream>>>(agg, feat_elems);
    magic_edge_kernel<<<edge_blocks, 256, 0, stream>>>(
        x, src, dst, edge_attr, Wa1, ba1, agg, n_edges);
    magic_node_kernel<<<node_blocks, 256, 0, stream>>>(
        agg, x, W1, b1, e1, h1, n_nodes, /*final_relu=*/0);

    // ---- layer 2 ----
    magic_zero_kernel<<<zero_blocks, 256, 0, stream>>>(agg, feat_elems);
    magic_edge_kernel<<<edge_blocks, 256, 0, stream>>>(
        h1, src, dst, edge_attr, Wa2, ba2, agg, n_edges);
    magic_node_kernel<<<node_blocks, 256, 0, stream>>>(
        agg, h1, W2, b2, e2, out, n_nodes, /*final_relu=*/1);
}